// EFDM_loss_14328010900127
// MI455X (gfx1250) — compile-verified
//
#include <hip/hip_runtime.h>
#include <hip/hip_bf16.h>
#include <stdint.h>

// Problem constants: L=2, P=2, B=4, C=128, H*W=4096, K=2
#define NROW 4096      // rows per tensor (S or T): 2L*P*B*C = 4*2*4*128
#define NEL  4096      // elements per row: H*W
#define HALF_ROWS 2048 // rows per single input array (L*P*B*C)
#define NGRP 1024      // phase-2 groups: (2L*P)*C = 8*128
#define CC   128
#define BB   4

typedef __attribute__((ext_vector_type(2))) float v2f;
typedef __attribute__((ext_vector_type(8))) float v8f;

// ---------------------------------------------------------------------------
// Kernel 1: sort one 4096-float row per workgroup (hybrid bitonic).
// Each thread owns a contiguous 16-element block in registers; substages with
// j<=8 run entirely in VGPRs (direction uniform per block for k>=32), only
// j>=16 substages go through LDS (36 LDS passes instead of 78).
// Input row is DMA'd into LDS with global_load_async_to_lds_b128 (ASYNCcnt);
// final sorted block is written register->global, fully coalesced.
// ---------------------------------------------------------------------------
__global__ __launch_bounds__(256) void efdm_sort_rows(
    const float* __restrict__ styleE, const float* __restrict__ styleS,
    const float* __restrict__ transE, const float* __restrict__ transS,
    float* __restrict__ ws)
{
    __shared__ __align__(16) float buf[NEL];
    const int row = blockIdx.x;          // 0..8191
    const int tid = threadIdx.x;

    const float* src;
    float* dst;
    if (row < NROW) {
        src = (row < HALF_ROWS) ? styleE + (size_t)row * NEL
                                : styleS + (size_t)(row - HALF_ROWS) * NEL;
        dst = ws + (size_t)row * NEL;
    } else {
        const int r = row - NROW;
        src = (r < HALF_ROWS) ? transE + (size_t)r * NEL
                              : transS + (size_t)(r - HALF_ROWS) * NEL;
        dst = ws + (size_t)NROW * NEL + (size_t)r * NEL;
    }

    // Async DMA global -> LDS: 4 x b128 per thread (16 KiB total per block).
    const unsigned lds_base = (unsigned)(uintptr_t)(&buf[0]); // low 32b = LDS offset
    #pragma unroll
    for (int c = 0; c < 4; ++c) {
        const unsigned off16 = (unsigned)(tid + c * 256) * 16u;
        const unsigned ldsa  = lds_base + off16;
        const uint64_t ga    = (uint64_t)(uintptr_t)src + (uint64_t)off16;
        asm volatile("global_load_async_to_lds_b128 %0, %1, off"
                     :
                     : "v"(ldsa), "v"(ga)
                     : "memory");
    }
    asm volatile("s_wait_asynccnt 0" ::: "memory");
    __syncthreads();

    const int base = tid * 16;
    float e[16];

    // ---- k = 2..16: full network inside each 16-block, in registers ----
    #pragma unroll
    for (int li = 0; li < 16; ++li) e[li] = buf[base + li];

    #pragma unroll
    for (int kk = 1; kk <= 4; ++kk) {            // k = 2,4,8,16
        const int k = 1 << kk;
        #pragma unroll
        for (int jj = kk - 1; jj >= 0; --jj) {   // j = k/2 .. 1
            const int j = 1 << jj;
            #pragma unroll
            for (int n = 0; n < 8; ++n) {
                const int li  = ((n & ~(j - 1)) << 1) | (n & (j - 1));
                const int lxj = li | j;
                const bool up = (((base + li) & k) == 0);
                const float a = e[li], b2 = e[lxj];
                if ((a > b2) == up) { e[li] = b2; e[lxj] = a; }
            }
        }
    }
    #pragma unroll
    for (int li = 0; li < 16; ++li) buf[base + li] = e[li];
    __syncthreads();

    // ---- k = 32..4096 ----
    for (int k = 32; k <= NEL; k <<= 1) {
        // LDS substages: j = k/2 .. 16 (partners cross 16-blocks)
        for (int j = k >> 1; j >= 16; j >>= 1) {
            #pragma unroll 1
            for (int n = tid; n < NEL / 2; n += 256) {
                const int i   = ((n & ~(j - 1)) << 1) | (n & (j - 1));
                const int ixj = i | j;
                const float a = buf[i], b2 = buf[ixj];
                const bool up = ((i & k) == 0);
                if ((a > b2) == up) { buf[i] = b2; buf[ixj] = a; }
            }
            __syncthreads();
        }
        // Register substages j = 8,4,2,1: direction uniform per 16-block.
        const bool up = ((base & k) == 0);
        #pragma unroll
        for (int li = 0; li < 16; ++li) e[li] = buf[base + li];
        #pragma unroll
        for (int jj = 3; jj >= 0; --jj) {
            const int j = 1 << jj;
            #pragma unroll
            for (int n = 0; n < 8; ++n) {
                const int li  = ((n & ~(j - 1)) << 1) | (n & (j - 1));
                const int lxj = li | j;
                const float a = e[li], b2 = e[lxj];
                if ((a > b2) == up) { e[li] = b2; e[lxj] = a; }
            }
        }
        if (k < NEL) {       // intermediate: back to LDS for next merge stage
            #pragma unroll
            for (int li = 0; li < 16; ++li) buf[base + li] = e[li];
            __syncthreads();
        }
    }

    // Final k=4096 result is in registers: store straight to global (64B/lane).
    #pragma unroll
    for (int q = 0; q < 4; ++q) {
        float4 v4 = make_float4(e[q * 4 + 0], e[q * 4 + 1],
                                e[q * 4 + 2], e[q * 4 + 3]);
        *(float4*)(dst + base + q * 4) = v4;
    }
}

// ---------------------------------------------------------------------------
// Kernel 2: per (l2p, c) group, 8x8 Gram of [T_b0..3 ; S_b0..3] sorted rows via
// V_WMMA_F32_16X16X4_F32. A row i == B col i == data row i, so one float2 load
// per lane feeds both operands (M=lane&15 selects row, lane>>4 selects K pair).
// Lanes m>=8 are zero-masked arithmetically so EXEC stays all-ones for WMMA.
// 4 waves split the K=4096 row length; fixed-order LDS reduction at the end.
// ---------------------------------------------------------------------------
__global__ __launch_bounds__(128) void efdm_gram(
    const float* __restrict__ ws, float* __restrict__ partials)
{
    const int g    = blockIdx.x;     // 0..1023
    const int l2p  = g >> 7;         // 0..7  (= l2*P + p)
    const int c    = g & 127;
    const int tid  = threadIdx.x;
    const int wave = tid >> 5;
    const int lane = tid & 31;
    const int m    = lane & 15;            // matrix row/col index
    const int koff = (lane >> 4) * 2;      // K offset within 4-wide step

    const float* baseS = ws;
    const float* baseT = ws + (size_t)NROW * NEL;

    // rows 0-3 = sorted T (b = m), rows 4-7 = sorted S (b = m-4), 8-15 zero
    const int bsel = m & 3;
    const size_t rowoff = (size_t)((l2p * BB + bsel) * CC + c) * NEL;
    const float* rowptr = ((m & 4) ? baseS : baseT) + rowoff;
    const float mask = (m < 8) ? 1.0f : 0.0f;

    v8f acc = {};
    const int r0 = wave * (NEL / 4);
    for (int r = r0; r < r0 + NEL / 4; r += 4) {
        v2f v = *(const v2f*)(rowptr + r + koff);
        v.x *= mask;
        v.y *= mask;
        // D = A*B + C ; A (16x4) and B (4x16) are the same 8 data rows.
        acc = __builtin_amdgcn_wmma_f32_16x16x4_f32(
            /*neg_a=*/false, v, /*neg_b=*/false, v,
            /*c_mod=*/(short)0, acc, /*reuse_a=*/false, /*reuse_b=*/false);
    }

    // Cross-wave reduction of the accumulator tile.
    __shared__ float red[4][256];
    #pragma unroll
    for (int vv = 0; vv < 8; ++vv) red[wave][lane * 8 + vv] = acc[vv];
    __syncthreads();

    // C/D layout: lane l (<16) holds G[M=vgpr][N=l]; we need M,N in 0..7.
    if (tid < 64) {
        const int mv = tid >> 3;  // M (vgpr index)
        const int nl = tid & 7;   // N (lane index)
        const float s = red[0][nl * 8 + mv] + red[1][nl * 8 + mv]
                      + red[2][nl * 8 + mv] + red[3][nl * 8 + mv];
        partials[(size_t)g * 64 + tid] = s;   // partials[g][M*8+N]
    }
}

// ---------------------------------------------------------------------------
// Kernel 3: deterministic final reduction over the 1024 group Grams + loss.
// G[b][b]=ΣT_b², G[4+b][4+b]=ΣS_b², G[b][4+b']=Σ sortT_b·sortS_b'.
// posd[b] = ΣT_b²+ΣS_b²-2G[b][4+b]; negd uses neg_idx; 1/(C*HW) cancels.
// ---------------------------------------------------------------------------
__global__ __launch_bounds__(64) void efdm_final(
    const float* __restrict__ partials, const int* __restrict__ neg_idx,
    float* __restrict__ out)
{
    __shared__ float G[64];
    const int t = threadIdx.x;
    float s = 0.0f;
    for (int g = 0; g < NGRP; ++g) s += partials[(size_t)g * 64 + t];
    G[t] = s;
    __syncthreads();

    if (t == 0) {
        float res = 0.0f;
        for (int b = 0; b < BB; ++b) {
            const float sqT = G[b * 8 + b];
            const float posd = sqT + G[(4 + b) * 8 + (4 + b)] - 2.0f * G[b * 8 + (4 + b)];
            float negd = 0.0f;
            for (int k = 0; k < 2; ++k) {
                const int nb = neg_idx[b * 2 + k];
                negd += sqT + G[(4 + nb) * 8 + (4 + nb)] - 2.0f * G[b * 8 + (4 + nb)];
            }
            res += posd / negd;
        }
        out[0] = res;
    }
}

// ---------------------------------------------------------------------------
// ws layout (floats): [0, NROW*NEL) sorted S | [NROW*NEL, 2*NROW*NEL) sorted T
//                     | [2*NROW*NEL, +NGRP*64) Gram partials   (~128.25 MiB)
// ---------------------------------------------------------------------------
extern "C" void kernel_launch(void* const* d_in, const int* in_sizes, int n_in,
                              void* d_out, int out_size, void* d_ws, size_t ws_size,
                              hipStream_t stream)
{
    (void)in_sizes; (void)n_in; (void)out_size; (void)ws_size;
    const float* styleE = (const float*)d_in[0];
    const float* styleS = (const float*)d_in[1];
    const float* transE = (const float*)d_in[2];
    const float* transS = (const float*)d_in[3];
    const int*   negidx = (const int*)d_in[4];

    float* ws       = (float*)d_ws;
    float* partials = ws + (size_t)2 * NROW * NEL;

    efdm_sort_rows<<<2 * NROW, 256, 0, stream>>>(styleE, styleS, transE, transS, ws);
    efdm_gram<<<NGRP, 128, 0, stream>>>(ws, partials);
    efdm_final<<<1, 64, 0, stream>>>(partials, negidx, (float*)d_out);
}